// MLPPredictor_84121229460231
// MI455X (gfx1250) — compile-verified
//
#include <hip/hip_runtime.h>
#include <hip/hip_bf16.h>
#include <cstdint>

typedef __attribute__((ext_vector_type(2))) float        v2f;
typedef __attribute__((ext_vector_type(4))) float        v4f;
typedef __attribute__((ext_vector_type(8))) float        v8f;
typedef __attribute__((ext_vector_type(4))) unsigned int v4u;
typedef __attribute__((ext_vector_type(4))) int          v4i;
typedef __attribute__((ext_vector_type(8))) int          v8i;

#define D_FEAT   256
#define HIDDEN   512
#define NCOLS    1024           // concat of A (512) and B (512) halves
#define KSTEPS   (D_FEAT / 4)   // 64
#define NTILES   (NCOLS / 16)   // 64
#define MROWS    32             // M rows per workgroup (2 x 16-row WMMA subtiles)
#define LDS_PITCH 260           // 256 + 4 pad -> conflict-free A-frag ds_load_b64

// ---------------------------------------------------------------------------
// Kernel 1: pack W1 into WMMA-B-fragment order.
// Wcat[k, n] = (n < 512) ? W1[n, k] : W1[n-512, 256+k]
// Bpack flat index o = (((ks*NTILES + ntile)*32 + lane)*2 + j)
//   lane <  16: holds Wcat[4*ks + j,     ntile*16 + lane]
//   lane >= 16: holds Wcat[4*ks + 2 + j, ntile*16 + lane - 16]
// ---------------------------------------------------------------------------
__global__ void pack_w_kernel(const float* __restrict__ W1, float* __restrict__ bpack) {
    int o = blockIdx.x * blockDim.x + threadIdx.x;
    const int total = KSTEPS * NTILES * 32 * 2;
    if (o >= total) return;
    int j     = o & 1;
    int lane  = (o >> 1) & 31;
    int ntile = (o >> 6) & (NTILES - 1);
    int ks    = o >> 12;
    int half  = lane >> 4;
    int k     = ks * 4 + half * 2 + j;
    int n     = ntile * 16 + (lane & 15);
    float v = (n < HIDDEN) ? W1[(size_t)n * (2 * D_FEAT) + k]
                           : W1[(size_t)(n - HIDDEN) * (2 * D_FEAT) + D_FEAT + k];
    bpack[o] = v;
}

// ---------------------------------------------------------------------------
// Kernel 2: Y[n_nodes, 1024] = h[n_nodes, 256] @ Wcat[256, 1024] via
// V_WMMA_F32_16X16X4_F32. One block (8 waves) per 32-row M tile; each wave
// computes a 32x128 strip (2 M-subtiles x 8 N-tiles of v8f accumulators),
// reusing every B fragment twice. The 32x256 h tile is DMA'd into LDS by the
// Tensor Data Mover with pad_enable producing the 260-float padded pitch.
// ---------------------------------------------------------------------------
__global__ void __launch_bounds__(256)
gemm_node_kernel(const float* __restrict__ h, const float* __restrict__ bpack,
                 float* __restrict__ Y, int n_nodes) {
    __shared__ float lds_h[MROWS * LDS_PITCH];   // 33,280 B

    const int tid = threadIdx.x;
    const int m0  = blockIdx.x * MROWS;

    if ((tid >> 5) == 0) {
        // ---- Tensor DMA descriptor (ISA 8.3 - 8.6) ----
        // 2D tensor: h is [n_nodes x 256] f32, tile = 256 x 32 at row m0.
        // Padding: after every 256 DWORDs (pad_interval=7) insert 4 DWORDs
        // (pad_amount=3) -> LDS row pitch 260 floats.
        unsigned lds_base = (unsigned)(uintptr_t)(&lds_h[0]);          // LDS byte addr = low 32 bits
        unsigned long long ga = (unsigned long long)(uintptr_t)(h + (size_t)m0 * D_FEAT);
        v4u g0;
        g0[0] = 1u;                                        // count=1 (valid), user mode
        g0[1] = lds_base;                                  // lds_addr
        g0[2] = (unsigned)(ga & 0xFFFFFFFFu);              // global_addr[31:0]
        g0[3] = (unsigned)((ga >> 32) & 0x01FFFFFFu)       // global_addr[56:32]
              | (2u << 30);                                // type = 2 ("image")
        v8i g1;
        g1[0] = (int)((2u << 16)        // data_size = 4 bytes
                    | (1u << 20)        // pad_enable
                    | (7u << 22)        // pad_interval: 256 DWORDs
                    | (3u << 25));      // pad_amount:   4 DWORDs
        g1[1] = (int)(((unsigned)D_FEAT & 0xFFFFu) << 16);               // tensor_dim0[15:0]
        g1[2] = (int)(((unsigned)D_FEAT >> 16)
                    | (((unsigned)n_nodes & 0xFFFFu) << 16));            // dim0 hi | tensor_dim1 lo
        g1[3] = (int)(((unsigned)n_nodes >> 16)
                    | ((unsigned)D_FEAT << 16));                         // dim1 hi | tile_dim0 = 256
        g1[4] = MROWS;                    // tile_dim1 = 32, tile_dim2 = 0 (unused)
        g1[5] = D_FEAT;                   // tensor_dim0_stride = 256 (lo32)
        g1[6] = 0;                        // stride hi | tensor_dim1_stride lo (unused)
        g1[7] = 0;
        v4i g2, g3;
        g2[0] = 1; g2[1] = 1; g2[2] = 0; g2[3] = 0;   // tensor_dim2=1, tensor_dim3=1 (benign)
        g3[0] = 0; g3[1] = 0; g3[2] = 1; g3[3] = 0;   // tensor_dim4=1 (bits 79:48), tile_dim4=0
        v8i g4 = {0, 0, 0, 0, 0, 0, 0, 0};            // unused trailing descriptor words
        __builtin_amdgcn_tensor_load_to_lds(g0, g1, g2, g3, g4, 0);
        __builtin_amdgcn_s_wait_tensorcnt(0);
    }
    __syncthreads();

    const int wave = tid >> 5;            // 0..7 -> N strip of 128 cols
    const int lane = tid & 31;
    const int half = lane >> 4;           // K-half selector for A/B frags
    const int lrow = lane & 15;           // M row (A) / N col (B,D) within tile
    const int nt0  = wave * 8;

    const v8f vzero = {0.f, 0.f, 0.f, 0.f, 0.f, 0.f, 0.f, 0.f};
    v8f acc[2][8];
    #pragma unroll
    for (int mt = 0; mt < 2; ++mt)
        #pragma unroll
        for (int t = 0; t < 8; ++t) acc[mt][t] = vzero;

    for (int ks = 0; ks < KSTEPS; ++ks) {
        // A fragments for the two 16-row M subtiles
        int k0 = ks * 4 + half * 2;
        v2f a0 = *(const v2f*)(&lds_h[lrow * LDS_PITCH + k0]);
        v2f a1 = *(const v2f*)(&lds_h[(lrow + 16) * LDS_PITCH + k0]);
        const float* bbase = bpack + (((size_t)ks * NTILES + nt0) * 32 + lane) * 2;
        #pragma unroll
        for (int t = 0; t < 8; ++t) {
            v2f b = *(const v2f*)(bbase + (size_t)t * 64);
            acc[0][t] = __builtin_amdgcn_wmma_f32_16x16x4_f32(
                false, a0, false, b, (short)0, acc[0][t], false, false);
            acc[1][t] = __builtin_amdgcn_wmma_f32_16x16x4_f32(
                false, a1, false, b, (short)0, acc[1][t], false, false);
        }
    }

    // Store D: VGPR q holds row M = q + 8*half, col N = lrow (ISA C/D layout)
    #pragma unroll
    for (int mt = 0; mt < 2; ++mt) {
        #pragma unroll
        for (int t = 0; t < 8; ++t) {
            int n = (nt0 + t) * 16 + lrow;
            float* yp = Y + (size_t)(m0 + mt * 16 + 8 * half) * NCOLS + n;
            #pragma unroll
            for (int q = 0; q < 8; ++q)
                yp[(size_t)q * NCOLS] = acc[mt][t][q];
        }
    }
}

// ---------------------------------------------------------------------------
// Kernel 3: per-edge score. One wave32 per edge:
//   score[e] = b2 + sum_j w2[j] * relu(Y[src][j] + Y[dst][512+j] + b1[j])
// ---------------------------------------------------------------------------
__global__ void __launch_bounds__(256)
edge_score_kernel(const float* __restrict__ Y, const int* __restrict__ src,
                  const int* __restrict__ dst, const float* __restrict__ b1,
                  const float* __restrict__ w2, const float* __restrict__ b2,
                  float* __restrict__ out, int n_edges) {
    int e    = (blockIdx.x * blockDim.x + threadIdx.x) >> 5;
    int lane = threadIdx.x & 31;
    if (e >= n_edges) return;

    const float* arow = Y + (size_t)src[e] * NCOLS;            // A half
    const float* brow = Y + (size_t)dst[e] * NCOLS + HIDDEN;   // B half

    float sum = 0.f;
    #pragma unroll
    for (int i = 0; i < 4; ++i) {
        int j = 4 * lane + 128 * i;
        v4f va = *(const v4f*)(arow + j);
        v4f vb = *(const v4f*)(brow + j);
        v4f vc = *(const v4f*)(b1 + j);
        v4f vw = *(const v4f*)(w2 + j);
        #pragma unroll
        for (int q = 0; q < 4; ++q) {
            float hid = va[q] + vb[q] + vc[q];
            hid = hid > 0.f ? hid : 0.f;
            sum = fmaf(hid, vw[q], sum);
        }
    }
    #pragma unroll
    for (int off = 16; off > 0; off >>= 1)
        sum += __shfl_xor(sum, off, 32);
    if (lane == 0) out[e] = sum + b2[0];
}

// ---------------------------------------------------------------------------
// Fallback (workspace too small or n_nodes % 32 != 0): direct per-edge MLP, one block/edge.
// ---------------------------------------------------------------------------
__global__ void __launch_bounds__(256)
edge_mlp_naive_kernel(const float* __restrict__ h, const int* __restrict__ src,
                      const int* __restrict__ dst, const float* __restrict__ W1,
                      const float* __restrict__ b1, const float* __restrict__ w2,
                      const float* __restrict__ b2, float* __restrict__ out,
                      int n_edges) {
    int e = blockIdx.x;
    if (e >= n_edges) return;
    __shared__ float he[2 * D_FEAT];
    __shared__ float partial[256];
    const float* hs = h + (size_t)src[e] * D_FEAT;
    const float* hd = h + (size_t)dst[e] * D_FEAT;
    int t = threadIdx.x;
    he[t]          = hs[t];
    he[t + D_FEAT] = hd[t];
    __syncthreads();
    float sum = 0.f;
    for (int jj = 0; jj < HIDDEN / 256; ++jj) {
        int jh = t + 256 * jj;
        float acc = b1[jh];
        const float* wr = W1 + (size_t)jh * (2 * D_FEAT);
        for (int d = 0; d < 2 * D_FEAT; ++d) acc = fmaf(wr[d], he[d], acc);
        acc = acc > 0.f ? acc : 0.f;
        sum = fmaf(acc, w2[jh], sum);
    }
    partial[t] = sum;
    __syncthreads();
    for (int s = 128; s > 0; s >>= 1) {
        if (t < s) partial[t] += partial[t + s];
        __syncthreads();
    }
    if (t == 0) out[e] = partial[0] + b2[0];
}

extern "C" void kernel_launch(void* const* d_in, const int* in_sizes, int n_in,
                              void* d_out, int out_size, void* d_ws, size_t ws_size,
                              hipStream_t stream) {
    const float* h   = (const float*)d_in[0];
    const int*   src = (const int*)d_in[1];
    const int*   dst = (const int*)d_in[2];
    const float* W1  = (const float*)d_in[3];
    const float* b1  = (const float*)d_in[4];
    const float* w2  = (const float*)d_in[5];
    const float* b2  = (const float*)d_in[6];
    float*       out = (float*)d_out;

    const int n_edges = in_sizes[1];
    const int n_nodes = in_sizes[0] / D_FEAT;

    const size_t y_bytes  = (size_t)n_nodes * NCOLS * sizeof(float);
    const size_t bp_elems = (size_t)KSTEPS * NTILES * 32 * 2;
    const size_t bp_bytes = bp_elems * sizeof(float);

    if (ws_size >= y_bytes + bp_bytes && (n_nodes % MROWS) == 0) {
        float* Y     = (float*)d_ws;
        float* bpack = (float*)((char*)d_ws + y_bytes);

        int pack_threads = (int)bp_elems;
        pack_w_kernel<<<(pack_threads + 255) / 256, 256, 0, stream>>>(W1, bpack);

        gemm_node_kernel<<<n_nodes / MROWS, 256, 0, stream>>>(h, bpack, Y, n_nodes);

        long long total_threads = (long long)n_edges * 32;
        int blocks = (int)((total_threads + 255) / 256);
        edge_score_kernel<<<blocks, 256, 0, stream>>>(Y, src, dst, b1, w2, b2, out, n_edges);
    } else {
        edge_mlp_naive_kernel<<<n_edges, 256, 0, stream>>>(h, src, dst, W1, b1, w2, b2,
                                                           out, n_edges);
    }
}